// TF_AwareBlock_DCN_67147518705702
// MI455X (gfx1250) — compile-verified
//
#include <hip/hip_runtime.h>
#include <hip/hip_bf16.h>
#include <math.h>

typedef float v2f __attribute__((ext_vector_type(2)));
typedef float v8f __attribute__((ext_vector_type(8)));
typedef unsigned int ui32x4 __attribute__((ext_vector_type(4)));
typedef int i32x4 __attribute__((ext_vector_type(4)));
typedef int i32x8 __attribute__((ext_vector_type(8)));

#define BN_EPS 1e-5f

// Problem constants
#define BATCH 16
#define CCH   256
#define HH    32
#define WW    32
#define PIX   1024
#define MID   1024
#define RD    256
#define K2    9
#define OMCH  27

// Workspace offsets (bytes)
#define OFF_X1     0ull
#define OFF_H1     16777216ull     // also aliases attn (16.8MB < 67MB)
#define OFF_H2     83886080ull     // also aliases om (1.77MB < 67MB)
#define OFF_SUMS   150994944ull
#define OFF_GATE   151060480ull
#define OFF_WDCNT  151126016ull    // w_dcn packed [9][128][256][2]
#define OFF_WPWT   153485312ull    // w_pw  packed [128][1024][2] (bn2-scaled)
#define OFF_WPWLT  154533888ull    // w_pwl packed [512][256][2]
#define OFF_BBPW   155582464ull    // folded bn2 bias through w_pw: [1024]

__device__ __forceinline__ v8f wmma4(v2f a, v2f b, v8f c) {
  // D(16x16,f32) = A(16x4,f32) x B(4x16,f32) + C  — CDNA5 V_WMMA_F32_16X16X4_F32
  return __builtin_amdgcn_wmma_f32_16x16x4_f32(false, a, false, b, (short)0, c, false, false);
}

__device__ __forceinline__ float sigmoidf(float x) { return 1.0f / (1.0f + expf(-x)); }

// TDM: async 2-D tile load (rows of 16 f32 pixels x `rows` lines, line stride in elems)
// Descriptor per CDNA5 ISA ch.8 (group0: count/lds/global/type, group1: dims/strides).
__device__ __forceinline__ void tdm_load_tile_2d(unsigned lds_addr, unsigned long long gaddr,
                                                 int rows, int stride_elems) {
  ui32x4 g0;
  g0[0] = 1u;                                                    // count=1, user mode
  g0[1] = lds_addr;                                              // LDS byte address
  g0[2] = (unsigned)(gaddr & 0xFFFFFFFFull);                     // global_addr[31:0]
  g0[3] = (unsigned)((gaddr >> 32) & 0x1FFFFFFull) | (2u << 30); // global_addr[56:32], type=2
  i32x8 g1;
  g1[0] = (int)(2u << 16);                 // data_size=2 (4B); no multicast/pad/iterate
  g1[1] = (int)(16u << 16);                // tensor_dim0 = 16 (bits 79:48 low half)
  g1[2] = (int)((unsigned)rows << 16);     // tensor_dim1 = rows (bits 111:80 low half)
  g1[3] = (int)(16u << 16);                // tile_dim0 = 16 (bits 127:112)
  g1[4] = rows;                            // tile_dim1 (bits 143:128)
  g1[5] = stride_elems;                    // tensor_dim0_stride[31:0]
  g1[6] = 0;
  g1[7] = 0;
  i32x4 gz4 = {0, 0, 0, 0};
  i32x8 gz8 = {0, 0, 0, 0, 0, 0, 0, 0};
  __builtin_amdgcn_tensor_load_to_lds(g0, g1, gz4, gz4, gz8, 0);
}

// ---------------------------------------------------------------- prep: packed weight transposes
__global__ void k_prep(const float* __restrict__ w_dcn, const float* __restrict__ w_pw,
                       const float* __restrict__ w_pwl, const float* __restrict__ g2,
                       const float* __restrict__ v2, float* __restrict__ wdT2,
                       float* __restrict__ wpwT2, float* __restrict__ wpwlT2) {
  int i = blockIdx.x * blockDim.x + threadIdx.x;
  if (i < K2 * CCH * CCH) {                       // wdT2[k][c>>1][d][c&1] = w_dcn[d][c][k]
    int j = i & 1, q = i >> 1;
    int d = q & 255, r = q >> 8;
    int c2 = r & 127, k = r >> 7;
    wdT2[i] = w_dcn[(d * CCH + 2 * c2 + j) * K2 + k];
  } else if (i < K2 * CCH * CCH + CCH * MID) {    // wpwT2[c>>1][oc][c&1] = w_pw[oc][c]*inv2[c]
    int i2 = i - K2 * CCH * CCH;
    int j = i2 & 1, q = i2 >> 1;
    int oc = q & 1023, c2 = q >> 10;
    int c = 2 * c2 + j;
    float inv = g2[c] * rsqrtf(v2[c] + BN_EPS);
    wpwT2[i2] = w_pw[oc * CCH + c] * inv;
  } else if (i < K2 * CCH * CCH + 2 * CCH * MID) {// wpwlT2[m>>1][oc][m&1] = w_pwl[oc][m]
    int i3 = i - K2 * CCH * CCH - CCH * MID;
    int j = i3 & 1, q = i3 >> 1;
    int oc = q & 255, m2 = q >> 8;
    wpwlT2[i3] = w_pwl[oc * MID + 2 * m2 + j];
  }
}

// ---------------------------------------------------------------- fold bn2 shift through w_pw
__global__ void k_fold(const float* __restrict__ w_pw, const float* __restrict__ g2,
                       const float* __restrict__ b2, const float* __restrict__ m2,
                       const float* __restrict__ v2, float* __restrict__ bbpw) {
  int oc = blockIdx.x * blockDim.x + threadIdx.x;
  if (oc >= MID) return;
  float s = 0.f;
  for (int c = 0; c < CCH; ++c) {
    float inv = g2[c] * rsqrtf(v2[c] + BN_EPS);
    s += w_pw[oc * CCH + c] * (b2[c] - m2[c] * inv);
  }
  bbpw[oc] = s;
}

// ---------------------------------------------------------------- bn1 -> attn
__global__ void k_bn1(const float* __restrict__ x, const float* __restrict__ g,
                      const float* __restrict__ bb, const float* __restrict__ m,
                      const float* __restrict__ v, float* __restrict__ attn) {
  int i = blockIdx.x * blockDim.x + threadIdx.x;
  if (i >= BATCH * CCH * PIX) return;
  int c = (i >> 10) & (CCH - 1);
  float inv = g[c] * rsqrtf(v[c] + BN_EPS);
  attn[i] = x[i] * inv + (bb[c] - m[c] * inv);
}

// ---------------------------------------------------------------- 3x3 conv 256->27 (+bias, sigmoid on mask)
__global__ void k_om(const float* __restrict__ attn, const float* __restrict__ w_om,
                     const float* __restrict__ b_om, float* __restrict__ om) {
  __shared__ float aRow[3 * 34];
  __shared__ float wTap[OMCH * K2];
  const int t = threadIdx.x;            // 288 threads
  const int wcol = t & 31;
  const int oc9 = t >> 5;               // 0..8
  const int h = blockIdx.x;
  const int b = blockIdx.y;
  float acc0 = 0.f, acc1 = 0.f, acc2 = 0.f;
  const float* aB = attn + (size_t)b * CCH * PIX;
  for (int c = 0; c < CCH; ++c) {
    __syncthreads();
    if (t < 102) {
      int rr = t / 34, cc = t % 34;
      int hh = h + rr - 1, ww2 = cc - 1;
      aRow[t] = (hh >= 0 && hh < HH && ww2 >= 0 && ww2 < WW)
                    ? aB[c * PIX + hh * WW + ww2] : 0.f;
    }
    if (t < OMCH * K2) {
      int oc = t / K2, tap = t % K2;
      wTap[t] = w_om[(oc * CCH + c) * K2 + tap];
    }
    __syncthreads();
#pragma unroll
    for (int tap = 0; tap < 9; ++tap) {
      int ky = tap / 3, kx = tap % 3;
      float av = aRow[ky * 34 + wcol + kx];
      acc0 += av * wTap[(oc9) * K2 + tap];
      acc1 += av * wTap[(oc9 + 9) * K2 + tap];
      acc2 += av * wTap[(oc9 + 18) * K2 + tap];
    }
  }
  size_t base = ((size_t)b * OMCH) * PIX + h * WW + wcol;
  om[base + (size_t)(oc9) * PIX]      = acc0 + b_om[oc9];
  om[base + (size_t)(oc9 + 9) * PIX]  = acc1 + b_om[oc9 + 9];
  om[base + (size_t)(oc9 + 18) * PIX] = sigmoidf(acc2 + b_om[oc9 + 18]);
}

// ---------------------------------------------------------------- deform sample + WMMA GEMM + bn_d + residual
__global__ void k_deform(const float* __restrict__ attn, const float* __restrict__ om,
                         const float* __restrict__ wdT2, const float* __restrict__ x,
                         const float* __restrict__ gd, const float* __restrict__ bd,
                         const float* __restrict__ md, const float* __restrict__ vd,
                         const float* __restrict__ ls1, float* __restrict__ x1) {
  __shared__ int   sIdx[144 * 4];
  __shared__ float sWt[144 * 4];
  __shared__ float sB[64 * 16];
  const int t = threadIdx.x;                  // 256
  const int p_base = blockIdx.x * 16;
  const int b = blockIdx.y;
  const float* omB = om + (size_t)b * OMCH * PIX;

  if (t < 144) {                              // bilinear params: 16 pixels x 9 taps
    int pl = t & 15, ktap = t >> 4;
    int p = p_base + pl;
    int h = p >> 5, w = p & 31;
    int ky = ktap / 3 - 1, kx = ktap % 3 - 1;
    float dy = omB[(size_t)(2 * ktap) * PIX + p];
    float dx = omB[(size_t)(2 * ktap + 1) * PIX + p];
    float mk = omB[(size_t)(18 + ktap) * PIX + p];
    float py = (float)(h + ky) + dy, px = (float)(w + kx) + dx;
    float fy = floorf(py), fx = floorf(px);
    float ly = py - fy, lx = px - fx;
    int y0 = (int)fy, x0 = (int)fx;
    int ys[4] = {y0, y0, y0 + 1, y0 + 1};
    int xs[4] = {x0, x0 + 1, x0, x0 + 1};
    float wg[4] = {(1.f - ly) * (1.f - lx), (1.f - ly) * lx, ly * (1.f - lx), ly * lx};
#pragma unroll
    for (int j = 0; j < 4; ++j) {
      int yi = ys[j], xi = xs[j];
      bool valid = (yi >= 0 && yi < HH && xi >= 0 && xi < WW);
      int yc = min(max(yi, 0), HH - 1), xc = min(max(xi, 0), WW - 1);
      sIdx[t * 4 + j] = yc * WW + xc;
      sWt[t * 4 + j] = valid ? wg[j] * mk : 0.f;
    }
  }

  const float* aB = attn + (size_t)b * CCH * PIX;
  const int lane = t & 31, wv = t >> 5;
  const int ln = lane & 15, hi = lane >> 4, k0 = hi * 2;
  const int d0 = wv * 16, d1 = wv * 16 + 128;
  v8f acc0 = {}, acc1 = {};

  for (int ktap = 0; ktap < K2; ++ktap) {
    for (int cb = 0; cb < CCH; cb += 64) {
      __syncthreads();
#pragma unroll
      for (int i = 0; i < 4; ++i) {           // stage s[c_chunk][16px] in LDS
        int v = t + i * 256;
        int cl = v >> 4, pl = v & 15;
        int pi = (ktap << 4) + pl;
        const float* ap = aB + (size_t)(cb + cl) * PIX;
        float s = sWt[pi * 4 + 0] * ap[sIdx[pi * 4 + 0]]
                + sWt[pi * 4 + 1] * ap[sIdx[pi * 4 + 1]]
                + sWt[pi * 4 + 2] * ap[sIdx[pi * 4 + 2]]
                + sWt[pi * 4 + 3] * ap[sIdx[pi * 4 + 3]];
        sB[cl * 16 + pl] = s;
      }
      __syncthreads();
      const float* wt2 = wdT2 + (size_t)ktap * (128 * 256 * 2);
#pragma unroll 4
      for (int kk = 0; kk < 64; kk += 4) {
        int cp = (cb + kk + k0) >> 1;         // (c even) pair index
        v2f bf; bf.x = sB[(kk + k0) * 16 + ln]; bf.y = sB[(kk + k0 + 1) * 16 + ln];
        v2f a0 = *(const v2f*)(wt2 + ((size_t)cp * 256 + d0 + ln) * 2);
        v2f a1 = *(const v2f*)(wt2 + ((size_t)cp * 256 + d1 + ln) * 2);
        acc0 = wmma4(a0, bf, acc0);
        acc1 = wmma4(a1, bf, acc1);
      }
    }
  }
#pragma unroll
  for (int r = 0; r < 8; ++r) {
    int p = p_base + ln;
    {
      int d = d0 + r + hi * 8;
      float inv = gd[d] * rsqrtf(vd[d] + BN_EPS);
      float beta = bd[d] - md[d] * inv;
      size_t idx = ((size_t)b * CCH + d) * PIX + p;
      x1[idx] = x[idx] + ls1[d] * (acc0[r] * inv + beta);
    }
    {
      int d = d1 + r + hi * 8;
      float inv = gd[d] * rsqrtf(vd[d] + BN_EPS);
      float beta = bd[d] - md[d] * inv;
      size_t idx = ((size_t)b * CCH + d) * PIX + p;
      x1[idx] = x[idx] + ls1[d] * (acc1[r] * inv + beta);
    }
  }
}

// ---------------------------------------------------------------- pw: TDM-staged x1, bn2 prefolded, bn_pw, relu
__global__ void k_pw(const float* __restrict__ x1, const float* __restrict__ wpwT2,
                     const float* __restrict__ bbpw,
                     const float* __restrict__ gp, const float* __restrict__ bp,
                     const float* __restrict__ mp, const float* __restrict__ vp,
                     float* __restrict__ h1) {
  __shared__ float tB[CCH * 16];
  const int t = threadIdx.x;                  // 256
  const int p_base = blockIdx.x * 16;
  const int b = blockIdx.z;
  if (t < 32) {                               // wave 0 issues the tensor DMA
    unsigned lds = (unsigned)(unsigned long long)(size_t)&tB[0];
    unsigned long long ga =
        (unsigned long long)(size_t)(x1 + ((size_t)b * CCH) * PIX + p_base);
    tdm_load_tile_2d(lds, ga, CCH, PIX);
    __builtin_amdgcn_s_wait_tensorcnt(0);
  }
  __syncthreads();
  const int lane = t & 31, ln = lane & 15, hi = lane >> 4, k0 = hi * 2;
  const int oc0 = blockIdx.y * 256 + (t >> 5) * 16;
  const int oc1 = oc0 + 128;
  v8f acc0 = {}, acc1 = {};
#pragma unroll 4
  for (int kk = 0; kk < CCH; kk += 4) {
    int cp = (kk + k0) >> 1;
    v2f bf; bf.x = tB[(kk + k0) * 16 + ln]; bf.y = tB[(kk + k0 + 1) * 16 + ln];
    v2f a0 = *(const v2f*)(wpwT2 + ((size_t)cp * MID + oc0 + ln) * 2);
    v2f a1 = *(const v2f*)(wpwT2 + ((size_t)cp * MID + oc1 + ln) * 2);
    acc0 = wmma4(a0, bf, acc0);
    acc1 = wmma4(a1, bf, acc1);
  }
#pragma unroll
  for (int r = 0; r < 8; ++r) {
    {
      int oc = oc0 + r + hi * 8;
      float inv = gp[oc] * rsqrtf(vp[oc] + BN_EPS);
      float hv = (acc0[r] + bbpw[oc]) * inv + (bp[oc] - mp[oc] * inv);
      h1[((size_t)b * MID + oc) * PIX + p_base + ln] = fmaxf(hv, 0.f);
    }
    {
      int oc = oc1 + r + hi * 8;
      float inv = gp[oc] * rsqrtf(vp[oc] + BN_EPS);
      float hv = (acc1[r] + bbpw[oc]) * inv + (bp[oc] - mp[oc] * inv);
      h1[((size_t)b * MID + oc) * PIX + p_base + ln] = fmaxf(hv, 0.f);
    }
  }
}

// ---------------------------------------------------------------- dw 7x7 depthwise + bn + relu + SE sums
__global__ void k_dw(const float* __restrict__ h1, const float* __restrict__ w_dw,
                     const float* __restrict__ gd, const float* __restrict__ bd,
                     const float* __restrict__ md, const float* __restrict__ vd,
                     float* __restrict__ h2, float* __restrict__ sums) {
  __shared__ float tile[38 * 38];
  __shared__ float wk[49];
  __shared__ float red[256];
  const int t = threadIdx.x;
  const int ch = blockIdx.x, b = blockIdx.y;
  const float* src = h1 + ((size_t)b * MID + ch) * PIX;
#pragma unroll
  for (int i = 0; i < 6; ++i) {
    int v = t + i * 256;
    if (v < 38 * 38) {
      int row = v / 38, col = v % 38;
      int hh = row - 3, ww2 = col - 3;
      tile[v] = (hh >= 0 && hh < HH && ww2 >= 0 && ww2 < WW) ? src[hh * WW + ww2] : 0.f;
    }
  }
  if (t < 49) wk[t] = w_dw[ch * 49 + t];
  __syncthreads();
  float inv = gd[ch] * rsqrtf(vd[ch] + BN_EPS);
  float beta = bd[ch] - md[ch] * inv;
  float lsum = 0.f;
#pragma unroll
  for (int i = 0; i < 4; ++i) {
    int p = t + i * 256;
    int hh = p >> 5, ww2 = p & 31;
    float acc = 0.f;
#pragma unroll
    for (int ky = 0; ky < 7; ++ky)
#pragma unroll
      for (int kx = 0; kx < 7; ++kx)
        acc += tile[(hh + ky) * 38 + ww2 + kx] * wk[ky * 7 + kx];
    float val = fmaxf(acc * inv + beta, 0.f);
    h2[((size_t)b * MID + ch) * PIX + p] = val;
    lsum += val;
  }
  red[t] = lsum;
  __syncthreads();
  for (int s = 128; s > 0; s >>= 1) {
    if (t < s) red[t] += red[t + s];
    __syncthreads();
  }
  if (t == 0) sums[b * MID + ch] = red[0];
}

// ---------------------------------------------------------------- SE MLP -> gate
__global__ void k_se(const float* __restrict__ sums, const float* __restrict__ w1,
                     const float* __restrict__ bb1, const float* __restrict__ w2,
                     const float* __restrict__ bb2, float* __restrict__ gate) {
  __shared__ float se[MID];
  __shared__ float hid[RD];
  const int t = threadIdx.x, b = blockIdx.x;
#pragma unroll
  for (int i = 0; i < 4; ++i) se[t + i * 256] = sums[b * MID + t + i * 256] * (1.f / (float)PIX);
  __syncthreads();
  float a = bb1[t];
  for (int m = 0; m < MID; ++m) a += se[m] * w1[(size_t)t * MID + m];
  hid[t] = fmaxf(a, 0.f);
  __syncthreads();
#pragma unroll
  for (int i = 0; i < 4; ++i) {
    int oc = t + i * 256;
    float g = bb2[oc];
    for (int r = 0; r < RD; ++r) g += hid[r] * w2[(size_t)oc * RD + r];
    gate[b * MID + oc] = sigmoidf(g);
  }
}

// ---------------------------------------------------------------- pwl: TDM-staged h2, gate on A-side, bn_pwl, ls2, residual
__global__ void k_pwl(const float* __restrict__ h2, const float* __restrict__ gate,
                      const float* __restrict__ wpwlT2, const float* __restrict__ gl,
                      const float* __restrict__ bl, const float* __restrict__ ml,
                      const float* __restrict__ vl, const float* __restrict__ ls2,
                      const float* __restrict__ x1, float* __restrict__ out) {
  __shared__ float tB[256 * 16];
  const int t = threadIdx.x;
  const int p_base = blockIdx.x * 16;
  const int b = blockIdx.y;
  const int lane = t & 31, ln = lane & 15, hi = lane >> 4, k0 = hi * 2;
  const int oc0 = (t >> 5) * 16, oc1 = oc0 + 128;
  const float* gateB = gate + (size_t)b * MID;
  v8f acc0 = {}, acc1 = {};
  for (int kc = 0; kc < 4; ++kc) {
    __syncthreads();                          // previous chunk fully consumed
    if (t < 32) {
      unsigned lds = (unsigned)(unsigned long long)(size_t)&tB[0];
      unsigned long long ga = (unsigned long long)(size_t)(
          h2 + ((size_t)b * MID + kc * 256) * PIX + p_base);
      tdm_load_tile_2d(lds, ga, 256, PIX);
      __builtin_amdgcn_s_wait_tensorcnt(0);
    }
    __syncthreads();
#pragma unroll 4
    for (int kk = 0; kk < 256; kk += 4) {
      int m = kc * 256 + kk + k0;             // even
      v2f gv; gv.x = gateB[m]; gv.y = gateB[m + 1];
      v2f bf; bf.x = tB[(kk + k0) * 16 + ln]; bf.y = tB[(kk + k0 + 1) * 16 + ln];
      v2f a0 = *(const v2f*)(wpwlT2 + ((size_t)(m >> 1) * CCH + oc0 + ln) * 2);
      v2f a1 = *(const v2f*)(wpwlT2 + ((size_t)(m >> 1) * CCH + oc1 + ln) * 2);
      a0 *= gv;
      a1 *= gv;
      acc0 = wmma4(a0, bf, acc0);
      acc1 = wmma4(a1, bf, acc1);
    }
  }
#pragma unroll
  for (int r = 0; r < 8; ++r) {
    int p = p_base + ln;
    {
      int oc = oc0 + r + hi * 8;
      float inv = gl[oc] * rsqrtf(vl[oc] + BN_EPS);
      float hv = acc0[r] * inv + (bl[oc] - ml[oc] * inv);
      size_t idx = ((size_t)b * CCH + oc) * PIX + p;
      out[idx] = x1[idx] + ls2[oc] * hv;
    }
    {
      int oc = oc1 + r + hi * 8;
      float inv = gl[oc] * rsqrtf(vl[oc] + BN_EPS);
      float hv = acc1[r] * inv + (bl[oc] - ml[oc] * inv);
      size_t idx = ((size_t)b * CCH + oc) * PIX + p;
      out[idx] = x1[idx] + ls2[oc] * hv;
    }
  }
}

extern "C" void kernel_launch(void* const* d_in, const int* in_sizes, int n_in,
                              void* d_out, int out_size, void* d_ws, size_t ws_size,
                              hipStream_t stream) {
  (void)in_sizes; (void)n_in; (void)out_size; (void)ws_size;
  const float* x     = (const float*)d_in[0];
  const float* g1    = (const float*)d_in[1];
  const float* b1    = (const float*)d_in[2];
  const float* m1    = (const float*)d_in[3];
  const float* v1    = (const float*)d_in[4];
  const float* w_om  = (const float*)d_in[5];
  const float* b_om  = (const float*)d_in[6];
  const float* w_dcn = (const float*)d_in[7];
  const float* gd    = (const float*)d_in[8];
  const float* bd    = (const float*)d_in[9];
  const float* md    = (const float*)d_in[10];
  const float* vd    = (const float*)d_in[11];
  const float* ls1   = (const float*)d_in[12];
  const float* g2    = (const float*)d_in[13];
  const float* b2    = (const float*)d_in[14];
  const float* m2    = (const float*)d_in[15];
  const float* v2    = (const float*)d_in[16];
  const float* w_pw  = (const float*)d_in[17];
  const float* g_pw  = (const float*)d_in[18];
  const float* b_pw  = (const float*)d_in[19];
  const float* m_pw  = (const float*)d_in[20];
  const float* v_pw  = (const float*)d_in[21];
  const float* w_dw  = (const float*)d_in[22];
  const float* g_dw  = (const float*)d_in[23];
  const float* b_dw  = (const float*)d_in[24];
  const float* m_dw  = (const float*)d_in[25];
  const float* v_dw  = (const float*)d_in[26];
  const float* w_se1 = (const float*)d_in[27];
  const float* b_se1 = (const float*)d_in[28];
  const float* w_se2 = (const float*)d_in[29];
  const float* b_se2 = (const float*)d_in[30];
  const float* w_pwl = (const float*)d_in[31];
  const float* g_pwl = (const float*)d_in[32];
  const float* b_pwl = (const float*)d_in[33];
  const float* m_pwl = (const float*)d_in[34];
  const float* v_pwl = (const float*)d_in[35];
  const float* ls2   = (const float*)d_in[36];

  char* ws = (char*)d_ws;
  float* x1v    = (float*)(ws + OFF_X1);
  float* attn   = (float*)(ws + OFF_H1);   // aliased: attn dies before h1 is written
  float* h1v    = (float*)(ws + OFF_H1);
  float* omv    = (float*)(ws + OFF_H2);   // aliased: om dies before h2 is written
  float* h2v    = (float*)(ws + OFF_H2);
  float* sums   = (float*)(ws + OFF_SUMS);
  float* gate   = (float*)(ws + OFF_GATE);
  float* wdT2   = (float*)(ws + OFF_WDCNT);
  float* wpwT2  = (float*)(ws + OFF_WPWT);
  float* wpwlT2 = (float*)(ws + OFF_WPWLT);
  float* bbpw   = (float*)(ws + OFF_BBPW);
  float* out    = (float*)d_out;

  {
    int n = K2 * CCH * CCH + 2 * CCH * MID;
    k_prep<<<(n + 255) / 256, 256, 0, stream>>>(w_dcn, w_pw, w_pwl, g2, v2,
                                                wdT2, wpwT2, wpwlT2);
  }
  k_fold<<<MID / 256, 256, 0, stream>>>(w_pw, g2, b2, m2, v2, bbpw);
  k_bn1<<<(BATCH * CCH * PIX + 255) / 256, 256, 0, stream>>>(x, g1, b1, m1, v1, attn);
  k_om<<<dim3(HH, BATCH), 288, 0, stream>>>(attn, w_om, b_om, omv);
  k_deform<<<dim3(PIX / 16, BATCH), 256, 0, stream>>>(attn, omv, wdT2, x, gd, bd, md, vd,
                                                      ls1, x1v);
  k_pw<<<dim3(PIX / 16, MID / 256, BATCH), 256, 0, stream>>>(x1v, wpwT2, bbpw,
                                                             g_pw, b_pw, m_pw, v_pw, h1v);
  k_dw<<<dim3(MID, BATCH), 256, 0, stream>>>(h1v, w_dw, g_dw, b_dw, m_dw, v_dw, h2v, sums);
  k_se<<<BATCH, 256, 0, stream>>>(sums, w_se1, b_se1, w_se2, b_se2, gate);
  k_pwl<<<dim3(PIX / 16, BATCH), 256, 0, stream>>>(h2v, gate, wpwlT2, g_pwl, b_pwl, m_pwl,
                                                   v_pwl, ls2, x1v, out);
}